// PrefillGPT2Attention_18322330485528
// MI455X (gfx1250) — compile-verified
//
#include <hip/hip_runtime.h>
#include <hip/hip_bf16.h>

// ---------------------------------------------------------------------------
// GPT-2 prefill attention on MI455X (gfx1250), FP32 via V_WMMA_F32_16X16X4_F32.
// - GEMMs: one wave32 per 16x64 output tile (A fragment reused across 4 WMMAs)
// - Attention: flash-style, K/V tiles double-buffered in LDS via
//   GLOBAL_LOAD_ASYNC_TO_LDS_B128 (ASYNCcnt) when available.
//
// FP32 WMMA VGPR layouts (CDNA5 ISA 7.12.2):
//   A 16x4 : lane l holds A[l%16][2*(l/16) + {0,1}]              (v2f)
//   B 4x16 : lane l holds B[2*(l/16) + {0,1}][l%16]              (v2f)
//   C/D    : VGPR v of lane l holds C[v + 8*(l/16)][l%16]        (v8f)
// ---------------------------------------------------------------------------

typedef __attribute__((ext_vector_type(2))) float v2f;
typedef __attribute__((ext_vector_type(8))) float v8f;
typedef __attribute__((ext_vector_type(4))) int   v4i;

#define S_LEN  2048
#define DMODEL 1024
#define NHEADS 16
#define HDIM   64
#define BATCH  2
#define MTOT   (BATCH * S_LEN)   // 4096 rows
#define KVPITCH 68               // LDS row pitch (floats): 16B-aligned, 68%64==4

#define AS1 __attribute__((address_space(1)))
#define AS3 __attribute__((address_space(3)))

#if __has_builtin(__builtin_amdgcn_global_load_async_to_lds_b128)
#define HAVE_ASYNC 1
#else
#define HAVE_ASYNC 0
#endif

__device__ __forceinline__ v8f wmma_f32(v2f a, v2f b, v8f c) {
  return __builtin_amdgcn_wmma_f32_16x16x4_f32(false, a, false, b,
                                               (short)0, c, false, false);
}

__device__ __forceinline__ void wait_async_all() {
#if HAVE_ASYNC
#if __has_builtin(__builtin_amdgcn_s_wait_asynccnt)
  __builtin_amdgcn_s_wait_asynccnt(0);
#else
  asm volatile("s_wait_asynccnt 0x0" ::: "memory");
#endif
#endif
}

__device__ __forceinline__ void wait_async_16() {
#if HAVE_ASYNC
#if __has_builtin(__builtin_amdgcn_s_wait_asynccnt)
  __builtin_amdgcn_s_wait_asynccnt(16);
#else
  asm volatile("s_wait_asynccnt 0x10" ::: "memory");
#endif
#endif
}

__device__ __forceinline__ void wait_ds0() {
#if __has_builtin(__builtin_amdgcn_s_wait_dscnt)
  __builtin_amdgcn_s_wait_dscnt(0);
#else
  asm volatile("s_wait_dscnt 0x0" ::: "memory");
#endif
}

// Copy one 16x64-float K/V tile (contiguous 4KB in [B,H,S,hd]) into LDS with
// KVPITCH row pitch. 256 chunks of 16B; 8 per lane.
__device__ __forceinline__ void tile_copy(float* lds, const float* gsrc, int lane) {
#pragma unroll
  for (int rep = 0; rep < 8; ++rep) {
    const int c = lane + 32 * rep;          // chunk id 0..255
    const int row = c >> 4;                 // 16 chunks per 64-float row
    const int col = (c & 15) * 4;           // float offset within row
    float* dst = lds + row * KVPITCH + col;
    const float* src = gsrc + c * 4;
#if HAVE_ASYNC
    __builtin_amdgcn_global_load_async_to_lds_b128(
        (AS1 v4i*)src, (AS3 v4i*)dst, 0, 0);
#else
    *(float4*)dst = *(const float4*)src;
#endif
  }
}

// ---------------------------------------------------------------------------
// Kernel 1: fused QKV projection.  qkv = hs[4096,1024] @ W[1024,3072] + b
// One wave per 16(M)x64(N) tile; a 64-wide tile spans exactly one head.
// Q -> workspace [B,H,S,hd]; K,V -> d_out slots [B,H,S,hd].
// ---------------------------------------------------------------------------
__global__ __launch_bounds__(32) void qkv_gemm(
    const float* __restrict__ hs, const float* __restrict__ w,
    const float* __restrict__ bias,
    float* __restrict__ qws, float* __restrict__ kout, float* __restrict__ vout) {
  const int lane = threadIdx.x;
  const int n0 = blockIdx.x * 64;            // 0..3008
  const int m0 = blockIdx.y * 16;            // 0..4080
  const int half = lane >> 4, lid = lane & 15, koff = half * 2;

  v8f acc[4] = {{}, {}, {}, {}};
  const float* arow = hs + (size_t)(m0 + lid) * DMODEL;
  for (int k = 0; k < DMODEL; k += 4) {
    const v2f a = *(const v2f*)(arow + k + koff);
    const float* w0 = w + (size_t)(k + koff) * (3 * DMODEL) + n0 + lid;
    const float* w1 = w0 + (3 * DMODEL);
#pragma unroll
    for (int t = 0; t < 4; ++t) {
      v2f b; b.x = w0[t * 16]; b.y = w1[t * 16];
      acc[t] = wmma_f32(a, b, acc[t]);
    }
  }

  const int which = n0 >> 10;                // 0=Q 1=K 2=V
  const int h     = (n0 & 1023) >> 6;        // head (64-wide tile = one head)
  float* dst = (which == 0) ? qws : (which == 1) ? kout : vout;

#pragma unroll
  for (int t = 0; t < 4; ++t) {
    const float bc = bias[n0 + t * 16 + lid];
    const int d = t * 16 + lid;
#pragma unroll
    for (int v = 0; v < 8; ++v) {
      const int row = m0 + v + half * 8;
      const int b = row >> 11, s = row & 2047;
      dst[((size_t)(b * NHEADS + h) * S_LEN + s) * HDIM + d] = acc[t][v] + bc;
    }
  }
}

// ---------------------------------------------------------------------------
// Kernel 2: flash-style causal attention per (b,h, 16-row q tile).
// K/V tiles double-buffered in LDS (async when available); online softmax;
// ctx -> workspace in [B,S,D] layout.
// ---------------------------------------------------------------------------
__global__ __launch_bounds__(32) void attn_fwd(
    const float* __restrict__ qws, const float* __restrict__ kmat,
    const float* __restrict__ vmat, const float* __restrict__ attn_bias,
    float* __restrict__ ctxws) {
  __shared__ float kbuf[2][16 * KVPITCH];
  __shared__ float vbuf[2][16 * KVPITCH];
  __shared__ float pl[16 * 17];

  const int lane = threadIdx.x;
  const int half = lane >> 4, lid = lane & 15, koff = half * 2;
  const int q0 = blockIdx.x * 16;
  const int bh = blockIdx.y;
  const int b  = bh >> 4, h = bh & 15;
  const size_t base = (size_t)bh * S_LEN * HDIM;
  const float* kbase = kmat + base;
  const float* vbase = vmat + base;

  // Preload Q tile into registers in A-operand layout.
  v2f qreg[16];
  {
    const float* qrow = qws + base + (size_t)(q0 + lid) * HDIM;
#pragma unroll
    for (int dd = 0; dd < 16; ++dd)
      qreg[dd] = *(const v2f*)(qrow + 4 * dd + koff);
  }

  v8f acc0 = {}, acc1 = {}, acc2 = {}, acc3 = {};
  float mi[8], li[8];
#pragma unroll
  for (int v = 0; v < 8; ++v) { mi[v] = -3.0e38f; li[v] = 0.0f; }

  const int jmax = q0 >> 4;

  // Prologue: stage tile 0.
  tile_copy(kbuf[0], kbase, lane);
  tile_copy(vbuf[0], vbase, lane);

  for (int j = 0; j <= jmax; ++j) {
    const int cur = j & 1;
    if (j < jmax) {
      wait_ds0();                             // WAR: prior reads of nxt done
      tile_copy(kbuf[cur ^ 1], kbase + (size_t)(j + 1) * 16 * HDIM, lane);
      tile_copy(vbuf[cur ^ 1], vbase + (size_t)(j + 1) * 16 * HDIM, lane);
      wait_async_16();                        // tile j landed; j+1 in flight
    } else {
      wait_async_all();
    }
    __syncthreads();

    // ---- scores = Q (16x64) . K_j^T (64x16): 16 chained WMMAs from LDS ----
    v8f s = {};
    const float* kr = &kbuf[cur][lid * KVPITCH];
#pragma unroll
    for (int dd = 0; dd < 16; ++dd) {
      v2f bb = *(const v2f*)(kr + 4 * dd + koff);   // ds_load_b64, no conflicts
      s = wmma_f32(qreg[dd], bb, s);
    }

    // ---- scale + learned bias + causal mask ----
    const int k0 = j * 16;
    const int kcol = k0 + lid;
    float pv[8];
#pragma unroll
    for (int v = 0; v < 8; ++v) {
      const int qrow = q0 + v + half * 8;
      float x = s[v] * 0.125f + attn_bias[(size_t)qrow * S_LEN + kcol];
      if (kcol > qrow) x = -1e9f;
      pv[v] = x;
    }

    // ---- online softmax: reductions across the 16 lanes of each half ----
    float corr[8];
#pragma unroll
    for (int v = 0; v < 8; ++v) {
      float rm = pv[v];
      rm = fmaxf(rm, __shfl_xor(rm, 1, 32));
      rm = fmaxf(rm, __shfl_xor(rm, 2, 32));
      rm = fmaxf(rm, __shfl_xor(rm, 4, 32));
      rm = fmaxf(rm, __shfl_xor(rm, 8, 32));
      const float mnew = fmaxf(mi[v], rm);
      const float p = __expf(pv[v] - mnew);
      float rs = p;
      rs += __shfl_xor(rs, 1, 32);
      rs += __shfl_xor(rs, 2, 32);
      rs += __shfl_xor(rs, 4, 32);
      rs += __shfl_xor(rs, 8, 32);
      const float c = __expf(mi[v] - mnew);
      li[v] = li[v] * c + rs;
      mi[v] = mnew;
      corr[v] = c;
      pv[v] = p;
    }
#pragma unroll
    for (int v = 0; v < 8; ++v) {
      acc0[v] *= corr[v]; acc1[v] *= corr[v];
      acc2[v] *= corr[v]; acc3[v] *= corr[v];
    }

    // ---- transpose P through LDS into A-operand layout ----
#pragma unroll
    for (int v = 0; v < 8; ++v)
      pl[(v + half * 8) * 17 + lid] = pv[v];
    __syncthreads();

    // ---- ctx += P (16x16) . V_j (16x64): 4 col tiles x 4 K-steps ----
    const float* vl = vbuf[cur];
#pragma unroll
    for (int kk = 0; kk < 4; ++kk) {
      v2f a;
      a.x = pl[lid * 17 + 4 * kk + koff];
      a.y = pl[lid * 17 + 4 * kk + koff + 1];
      const int r0 = 4 * kk + koff;
      v2f b0, b1, b2, b3;
      b0.x = vl[r0 * KVPITCH +  0 + lid]; b0.y = vl[(r0 + 1) * KVPITCH +  0 + lid];
      b1.x = vl[r0 * KVPITCH + 16 + lid]; b1.y = vl[(r0 + 1) * KVPITCH + 16 + lid];
      b2.x = vl[r0 * KVPITCH + 32 + lid]; b2.y = vl[(r0 + 1) * KVPITCH + 32 + lid];
      b3.x = vl[r0 * KVPITCH + 48 + lid]; b3.y = vl[(r0 + 1) * KVPITCH + 48 + lid];
      acc0 = wmma_f32(a, b0, acc0);
      acc1 = wmma_f32(a, b1, acc1);
      acc2 = wmma_f32(a, b2, acc2);
      acc3 = wmma_f32(a, b3, acc3);
    }
    __syncthreads();
  }

  // ---- finalize: ctx/li -> workspace in [B,S,D] layout ----
#pragma unroll
  for (int v = 0; v < 8; ++v) {
    const int qrow = q0 + v + half * 8;
    const float inv = 1.0f / li[v];
    const size_t o = ((size_t)(b * S_LEN + qrow)) * DMODEL + h * HDIM;
    ctxws[o +  0 + lid] = acc0[v] * inv;
    ctxws[o + 16 + lid] = acc1[v] * inv;
    ctxws[o + 32 + lid] = acc2[v] * inv;
    ctxws[o + 48 + lid] = acc3[v] * inv;
  }
}

// ---------------------------------------------------------------------------
// Kernel 3: output projection. out = ctx[4096,1024] @ Wp[1024,1024] + bp
// One wave per 16x64 tile.
// ---------------------------------------------------------------------------
__global__ __launch_bounds__(32) void proj_gemm(
    const float* __restrict__ x, const float* __restrict__ w,
    const float* __restrict__ bias, float* __restrict__ out) {
  const int lane = threadIdx.x;
  const int n0 = blockIdx.x * 64;
  const int m0 = blockIdx.y * 16;
  const int half = lane >> 4, lid = lane & 15, koff = half * 2;

  v8f acc[4] = {{}, {}, {}, {}};
  const float* arow = x + (size_t)(m0 + lid) * DMODEL;
  for (int k = 0; k < DMODEL; k += 4) {
    const v2f a = *(const v2f*)(arow + k + koff);
    const float* w0 = w + (size_t)(k + koff) * DMODEL + n0 + lid;
    const float* w1 = w0 + DMODEL;
#pragma unroll
    for (int t = 0; t < 4; ++t) {
      v2f b; b.x = w0[t * 16]; b.y = w1[t * 16];
      acc[t] = wmma_f32(a, b, acc[t]);
    }
  }
#pragma unroll
  for (int t = 0; t < 4; ++t) {
    const float bc = bias[n0 + t * 16 + lid];
#pragma unroll
    for (int v = 0; v < 8; ++v) {
      const int row = m0 + v + half * 8;
      out[(size_t)row * DMODEL + n0 + t * 16 + lid] = acc[t][v] + bc;
    }
  }
}

// ---------------------------------------------------------------------------
extern "C" void kernel_launch(void* const* d_in, const int* in_sizes, int n_in,
                              void* d_out, int out_size, void* d_ws, size_t ws_size,
                              hipStream_t stream) {
  const float* hs        = (const float*)d_in[0];
  const float* c_attn_w  = (const float*)d_in[1];
  const float* c_attn_b  = (const float*)d_in[2];
  const float* attn_bias = (const float*)d_in[3];
  const float* c_proj_w  = (const float*)d_in[4];
  const float* c_proj_b  = (const float*)d_in[5];

  float* out  = (float*)d_out;                       // [B,S,D]
  float* kout = out  + (size_t)MTOT * DMODEL;        // [B,H,S,hd]
  float* vout = kout + (size_t)MTOT * DMODEL;        // [B,H,S,hd]

  float* qws   = (float*)d_ws;                       // [B,H,S,hd]  16.8 MB
  float* ctxws = qws + (size_t)MTOT * DMODEL;        // [B,S,D]     16.8 MB

  dim3 blk(32);
  qkv_gemm <<<dim3((3 * DMODEL) / 64, MTOT / 16), blk, 0, stream>>>(
      hs, c_attn_w, c_attn_b, qws, kout, vout);
  attn_fwd <<<dim3(S_LEN / 16, BATCH * NHEADS),   blk, 0, stream>>>(
      qws, kout, vout, attn_bias, ctxws);
  proj_gemm<<<dim3(DMODEL / 64, MTOT / 16),       blk, 0, stream>>>(
      ctxws, c_proj_w, c_proj_b, out);
}